// PiecewiseHawkesIntensity_18769007083726
// MI455X (gfx1250) — compile-verified
//
#include <hip/hip_runtime.h>
#include <stdint.h>

// Problem constants (from the reference file)
#define B_  16
#define P_  32
#define L_  512
#define M_  32
#define LE_ 2048

#define CHUNK_M  8           // heads staged per chunk (double-buffered)
#define NCHUNKS  4           // M_/CHUNK_M
#define NTHREADS 256         // 8 wave32 waves per block

typedef float v4f __attribute__((ext_vector_type(4)));
typedef int   v4i __attribute__((ext_vector_type(4)));

// AS-qualified pointee types for the gfx1250 async-LDS builtin:
//   void __builtin_amdgcn_global_load_async_to_lds_b128(v4i AS1*, v4i AS3*, imm, imm)
typedef __attribute__((address_space(1))) v4i g_v4i;
typedef __attribute__((address_space(3))) v4i l_v4i;

#if defined(__HIP_DEVICE_COMPILE__) && __has_builtin(__builtin_amdgcn_global_load_async_to_lds_b128)
#define ASYNC_LDS_BUILTIN 1
#else
#define ASYNC_LDS_BUILTIN 0
#endif

__device__ __forceinline__ void async_copy16(const float* __restrict__ g,
                                             float* __restrict__ l) {
#if ASYNC_LDS_BUILTIN
  // (global src, lds dst, imm offset, imm cpol) -> GLOBAL_LOAD_ASYNC_TO_LDS_B128
  __builtin_amdgcn_global_load_async_to_lds_b128(
      (g_v4i*)(void*)g, (l_v4i*)l, 0, 0);
#else
  unsigned lds_off = (unsigned)(uintptr_t)(__attribute__((address_space(3))) char*)l;
  unsigned long long ga = (unsigned long long)(uintptr_t)g;
  asm volatile("global_load_async_to_lds_b128 %0, %1, off"
               :: "v"(lds_off), "v"(ga)
               : "memory");
#endif
}

__device__ __forceinline__ void wait_async0() {
#if defined(__HIP_DEVICE_COMPILE__) && __has_builtin(__builtin_amdgcn_s_wait_asynccnt)
  __builtin_amdgcn_s_wait_asynccnt(0);
#elif defined(__HIP_DEVICE_COMPILE__)
  asm volatile("s_wait_asynccnt 0x0" ::: "memory");
#endif
}

// ---------------------------------------------------------------------------
// One block per (b,p).  LDS: event row (2KB) + double-buffered 8-head slabs
// (2 x 48KB) = 98KB of the WGP's 320KB.  Async staging of chunk c+1 overlaps
// the LDS-gather compute of chunk c.
// ---------------------------------------------------------------------------
__global__ __launch_bounds__(NTHREADS)
void hawkes_intensity_kernel(const float* __restrict__ qt,   // [B,P,LE]
                             const float* __restrict__ ev,   // [B,P,L]
                             const float* __restrict__ mu,   // [B,M,P,L]
                             const float* __restrict__ al,   // [B,M,P,L]
                             const float* __restrict__ be,   // [B,M,P,L]
                             const float* __restrict__ nc,   // [B]
                             float* __restrict__ out)        // [B,M,P,LE]
{
  __shared__ float s_ev[L_];
  __shared__ float s_mu[2][CHUNK_M * L_];
  __shared__ float s_al[2][CHUNK_M * L_];
  __shared__ float s_be[2][CHUNK_M * L_];

  const int tid = threadIdx.x;
  const int bp  = blockIdx.x;
  const int b   = bp >> 5;          // P_ == 32
  const int p   = bp & (P_ - 1);

  const float ncb    = nc[b];
  const float inv_nc = 1.0f / ncb;

  const float* ev_row = ev + (size_t)(b * P_ + p) * L_;
  const float* q_row  = qt + (size_t)(b * P_ + p) * LE_;

  const size_t par0    = ((size_t)b * M_ * P_ + p) * (size_t)L_; // m=0 row start
  const size_t mstride = (size_t)P_ * L_;                        // stride between heads

  // ---- stage event-time row (128 x b128) ---------------------------------
  for (int k = tid; k < L_ / 4; k += NTHREADS)
    async_copy16(ev_row + k * 4, s_ev + k * 4);
  wait_async0();
  __syncthreads();

  // ---- async stage one 8-head chunk of mu/alpha/beta ---------------------
  auto stage_chunk = [&](int ci) {
    float* dmu = s_mu[ci & 1];
    float* dal = s_al[ci & 1];
    float* dbe = s_be[ci & 1];
    const size_t base = par0 + (size_t)(ci * CHUNK_M) * mstride;
    for (int k = tid; k < CHUNK_M * (L_ / 4); k += NTHREADS) { // 4 iterations
      const int    mm   = k >> 7;           // L_/4 == 128
      const int    c4   = (k & 127) * 4;
      const int    loff = mm * L_ + c4;
      const size_t goff = base + (size_t)mm * mstride + (size_t)c4;
      async_copy16(mu + goff, dmu + loff);
      async_copy16(al + goff, dal + loff);
      async_copy16(be + goff, dbe + loff);
    }
  };

  // Kick off chunk 0 now; it overlaps the binary search below.
  stage_chunk(0);

  // ---- per-thread: 8 consecutive queries; branchless lower_bound in LDS --
  int   idxc[8];
  float dtv[8];
  {
    const v4f q0 = __builtin_nontemporal_load((const v4f*)(q_row + tid * 8));
    const v4f q1 = __builtin_nontemporal_load((const v4f*)(q_row + tid * 8 + 4));
    float qn_[8];
#pragma unroll
    for (int j = 0; j < 4; ++j) {
      qn_[j]     = q0[j] / ncb;   // precise div: matches reference boundary ties
      qn_[j + 4] = q1[j] / ncb;
    }
#pragma unroll
    for (int j = 0; j < 8; ++j) {
      const float qn = qn_[j];
      // lower_bound over 512 sorted floats: pos = #elements < qn (side="left")
      unsigned pos = 0;
#pragma unroll
      for (int sh = 8; sh >= 0; --sh) {
        const unsigned cand = pos + (1u << sh);
        pos = (s_ev[cand - 1] < qn) ? cand : pos;
      }
      if (s_ev[pos] < qn) ++pos;          // saturated all-less case -> 512
      const int last = (int)pos - 1;      // last event strictly before qn
      const float tl = (last < 0) ? 0.0f : s_ev[last];
      idxc[j] = (last < 0) ? 0 : last;    // clamp(min=0) for param gather
      dtv[j]  = qn - tl;
    }
  }

  wait_async0();      // chunk 0 landed (per-wave), ...
  __syncthreads();    // ... and visible to all waves

  // ---- software pipeline over 4 chunks of 8 heads ------------------------
#pragma unroll 1
  for (int c = 0; c < NCHUNKS; ++c) {
    if (c + 1 < NCHUNKS)
      stage_chunk(c + 1);               // fills buf[(c+1)&1] while we compute

    const float* cmu = s_mu[c & 1];
    const float* cal = s_al[c & 1];
    const float* cbe = s_be[c & 1];

#pragma unroll 1
    for (int mm = 0; mm < CHUNK_M; ++mm) {
      const int m = c * CHUNK_M + mm;
      float* orow = out + ((size_t)(b * M_ + m) * P_ + p) * (size_t)LE_ + tid * 8;
      v4f r0, r1;
#pragma unroll
      for (int j = 0; j < 8; ++j) {
        const int   i   = mm * L_ + idxc[j];
        const float muv = cmu[i];
        const float alv = cal[i];
        const float bev = cbe[i];
        const float e   = __expf(-bev * dtv[j]);
        const float r   = (muv + (alv - muv) * e) * inv_nc;
        if (j < 4) r0[j] = r; else r1[j - 4] = r;
      }
      // write-once streaming output: non-temporal b128 stores
      __builtin_nontemporal_store(r0, (v4f*)orow);
      __builtin_nontemporal_store(r1, (v4f*)(orow + 4));
    }

    // Own chunk-(c+1) asyncs done (no-op on last iter), then one barrier:
    // guarantees (a) next chunk visible to all waves, (b) everyone finished
    // reading buf[c&1] before it is re-staged at the start of the next iteration.
    wait_async0();
    __syncthreads();
  }
}

// ---------------------------------------------------------------------------
extern "C" void kernel_launch(void* const* d_in, const int* in_sizes, int n_in,
                              void* d_out, int out_size, void* d_ws, size_t ws_size,
                              hipStream_t stream) {
  (void)in_sizes; (void)n_in; (void)out_size; (void)d_ws; (void)ws_size;
  const float* qt = (const float*)d_in[0];   // query_times  [B,P,LE]
  const float* ev = (const float*)d_in[1];   // event_times  [B,P,L]
  const float* mu = (const float*)d_in[2];   // mu           [B,M,P,L]
  const float* al = (const float*)d_in[3];   // alpha        [B,M,P,L]
  const float* be = (const float*)d_in[4];   // beta         [B,M,P,L]
  const float* nc = (const float*)d_in[5];   // norm_constants [B]
  float* out = (float*)d_out;                // [B,M,P,LE]

  dim3 grid(B_ * P_);
  dim3 block(NTHREADS);
  hipLaunchKernelGGL(hawkes_intensity_kernel, grid, block, 0, stream,
                     qt, ev, mu, al, be, nc, out);
}